// DigitCaps_56556129354671
// MI455X (gfx1250) — compile-verified
//
#include <hip/hip_runtime.h>
#include <hip/hip_bf16.h>

// DigitCaps dynamic routing, fully fused (u_hat never materialized to memory).
// Shapes: x[256,1152,8] f32, W[1152,10,16,8] f32, out v[256,10,16] f32.
//
//  iter0: s0 = (1/10) * GEMM([256x9216] @ [9216x160]) via v_wmma_f32_16x16x4_f32
//         A tile staged in LDS (coalesced, bank-conflict-free), reused across
//         all 10 output capsules (10 accumulators / block), split-K=36 with
//         native global_atomic_add_f32 reduction.
//  iter t>=1: per (b,i) recompute u_hat[10][16] in VGPRs, logits = running dots
//             with stored v_0..v_{t-1}, softmax over o, ds_add_f32 block
//             reduction + one global f32 atomic per element.
//  squash(s) -> v_t (final -> d_out).
// Workspace: s[40960] + v0[40960] + v1[40960] floats = 480 KB.

typedef float v2f __attribute__((ext_vector_type(2)));
typedef float v8f __attribute__((ext_vector_type(8)));

#define B_SZ      256
#define IN_CAPS   1152
#define IN_DIM    8
#define OUT_CAPS  10
#define OUT_DIM   16
#define KTOT      (IN_CAPS * IN_DIM)          // 9216
#define SV_ELEMS  (B_SZ * OUT_CAPS * OUT_DIM) // 40960
#define KSPLIT    36                          // 36 * 256 K = 9216
#define ROWP      260                         // padded LDS row (floats)

__global__ void zero_kernel(float* __restrict__ p, int n) {
    int i = blockIdx.x * blockDim.x + threadIdx.x;
    if (i < n) p[i] = 0.0f;
}

// Iteration-0 GEMM: s[b, o*16+d] += sum_K x[b,K] * W[((i*10+o)*16+d)*8+k],
// K = i*8+k. Grid: (16 M-tiles, 36 K-chunks). One wave32 per block;
// each K-step: 1 LDS A-fragment feeds 10 WMMAs (one per output capsule).
__global__ __launch_bounds__(32) void s0_gemm_wmma(const float* __restrict__ x,
                                                   const float* __restrict__ W,
                                                   float* __restrict__ s) {
    __shared__ float xtile[16 * ROWP];        // 16 rows x 256 K, rows padded
    const int lane = threadIdx.x;
    const int half = lane >> 4;               // 0: K={0,1}, 1: K={2,3}
    const int r    = lane & 15;
    const int m0   = blockIdx.x << 4;         // batch tile base
    const int K0   = blockIdx.y << 8;         // K-chunk base (32 input caps)

    // Cooperative coalesced stage of x[m0..m0+15][K0..K0+255] into LDS.
    // 1024 float4 total; 32 lanes x 32 steps; 64 float4 per row.
    #pragma unroll
    for (int j = 0; j < 32; ++j) {
        const int f   = (j << 5) + lane;      // float4 index in tile
        const int row = f >> 6;
        const int c4  = f & 63;
        const float4 val = *((const float4*)(x + (m0 + row) * KTOT + K0) + c4);
        *(float4*)&xtile[row * ROWP + (c4 << 2)] = val;
    }
    __syncthreads();

    v8f acc[OUT_CAPS] = {};

    #pragma unroll 2
    for (int kk = 0; kk < 64; ++kk) {
        const int koff = (kk << 2) + (half << 1);     // lane's K offset (even)
        // A fragment from LDS: row r, K = koff, koff+1. Bank-conflict-free.
        v2f a = *(const v2f*)&xtile[r * ROWP + koff];
        // B fragments: column n = o*16 + r, rows Ka, Ka+1 (k-contiguous in W)
        const int Ka = K0 + koff;
        const int i  = Ka >> 3;
        const int k  = Ka & 7;
        const float* wb = W + i * (OUT_CAPS * OUT_DIM * IN_DIM) + (r << 3) + k;
        #pragma unroll
        for (int o = 0; o < OUT_CAPS; ++o) {
            v2f b = *(const v2f*)(wb + (o << 7));     // o stride = 128 floats
            acc[o] = __builtin_amdgcn_wmma_f32_16x16x4_f32(
                false, a, false, b, (short)0, acc[o], false, false);
        }
    }

    // D layout: VGPR v -> M = v + 8*half, N = r. Accumulate across K-chunks.
    #pragma unroll
    for (int o = 0; o < OUT_CAPS; ++o) {
        #pragma unroll
        for (int v = 0; v < 8; ++v) {
            const int m = m0 + v + (half << 3);
            __hip_atomic_fetch_add(&s[m * (OUT_CAPS * OUT_DIM) + (o << 4) + r],
                                   acc[o][v], __ATOMIC_RELAXED,
                                   __HIP_MEMORY_SCOPE_AGENT);
        }
    }
}

// Routing iteration t>=1. One lane per (b, i). vs = [nv][256,10,16] stored v's.
__global__ __launch_bounds__(128) void routing_iter(const float* __restrict__ x,
                                                    const float* __restrict__ W,
                                                    const float* __restrict__ vs,
                                                    int nv,
                                                    float* __restrict__ s) {
    __shared__ float ssh[OUT_CAPS * OUT_DIM];
    const int b = blockIdx.x;
    const int i = blockIdx.y * 128 + threadIdx.x;

    for (int t = threadIdx.x; t < OUT_CAPS * OUT_DIM; t += 128) ssh[t] = 0.0f;
    __syncthreads();

    float xi[IN_DIM];
    const float* xp = x + (b * IN_CAPS + i) * IN_DIM;
    #pragma unroll
    for (int k = 0; k < IN_DIM; ++k) xi[k] = xp[k];

    // Recompute u_hat[b,i,:,:] into VGPRs (W[i] block is 1280 contiguous floats)
    float uh[OUT_CAPS][OUT_DIM];
    const float* Wi = W + i * (OUT_CAPS * OUT_DIM * IN_DIM);
    #pragma unroll
    for (int o = 0; o < OUT_CAPS; ++o) {
        #pragma unroll
        for (int d = 0; d < OUT_DIM; ++d) {
            const float* w = Wi + ((o << 4) + d) * IN_DIM;
            float a = 0.0f;
            #pragma unroll
            for (int k = 0; k < IN_DIM; ++k) a = fmaf(w[k], xi[k], a);
            uh[o][d] = a;
        }
    }

    // Logits: running sum of dot(u_hat, v_t) over stored v's (b starts at 0)
    float L[OUT_CAPS];
    #pragma unroll
    for (int o = 0; o < OUT_CAPS; ++o) L[o] = 0.0f;
    for (int t = 0; t < nv; ++t) {
        const float* v = vs + t * SV_ELEMS + b * (OUT_CAPS * OUT_DIM);
        #pragma unroll
        for (int o = 0; o < OUT_CAPS; ++o) {
            float a = 0.0f;
            #pragma unroll
            for (int d = 0; d < OUT_DIM; ++d) a = fmaf(uh[o][d], v[(o << 4) + d], a);
            L[o] += a;
        }
    }

    // Softmax over o (10-wide, in registers)
    float mx = L[0];
    #pragma unroll
    for (int o = 1; o < OUT_CAPS; ++o) mx = fmaxf(mx, L[o]);
    float c[OUT_CAPS], sum = 0.0f;
    #pragma unroll
    for (int o = 0; o < OUT_CAPS; ++o) { c[o] = __expf(L[o] - mx); sum += c[o]; }
    const float inv = 1.0f / sum;

    // Accumulate c[o]*u_hat into LDS (ds_add_f32), then one global atomic each
    #pragma unroll
    for (int o = 0; o < OUT_CAPS; ++o) {
        const float co = c[o] * inv;
        #pragma unroll
        for (int d = 0; d < OUT_DIM; ++d)
            atomicAdd(&ssh[(o << 4) + d], co * uh[o][d]);
    }
    __syncthreads();
    for (int t = threadIdx.x; t < OUT_CAPS * OUT_DIM; t += 128)
        __hip_atomic_fetch_add(&s[b * (OUT_CAPS * OUT_DIM) + t], ssh[t],
                               __ATOMIC_RELAXED, __HIP_MEMORY_SCOPE_AGENT);
}

// v = squash(scale * s); one thread per (b,o) capsule vector of 16.
__global__ __launch_bounds__(256) void squash_kernel(const float* __restrict__ s,
                                                     float* __restrict__ v,
                                                     float scale) {
    const int idx = blockIdx.x * 256 + threadIdx.x;   // (b*10 + o)
    if (idx >= B_SZ * OUT_CAPS) return;
    const float* sp = s + idx * OUT_DIM;
    float t[OUT_DIM];
    float ns = 0.0f;
    #pragma unroll
    for (int d = 0; d < OUT_DIM; ++d) { t[d] = sp[d] * scale; ns = fmaf(t[d], t[d], ns); }
    const float f = (ns / (1.0f + ns)) * rsqrtf(ns + 1e-8f);
    float* vp = v + idx * OUT_DIM;
    #pragma unroll
    for (int d = 0; d < OUT_DIM; ++d) vp[d] = f * t[d];
}

extern "C" void kernel_launch(void* const* d_in, const int* in_sizes, int n_in,
                              void* d_out, int out_size, void* d_ws, size_t ws_size,
                              hipStream_t stream) {
    const float* x = (const float*)d_in[0];
    const float* W = (const float*)d_in[1];
    float* out = (float*)d_out;

    float* s  = (float*)d_ws;      // [256,10,16]
    float* v0 = s  + SV_ELEMS;     // stored v from iter 0
    float* v1 = v0 + SV_ELEMS;     // stored v from iter 1 (contiguous after v0)

    const dim3 gemm_grid(B_SZ / 16, KSPLIT);
    const dim3 rout_grid(B_SZ, IN_CAPS / 128);

    // iter 0: uniform coupling -> pure WMMA GEMM, scale 1/10 folded into squash
    zero_kernel<<<(SV_ELEMS + 255) / 256, 256, 0, stream>>>(s, SV_ELEMS);
    s0_gemm_wmma<<<gemm_grid, 32, 0, stream>>>(x, W, s);
    squash_kernel<<<(B_SZ * OUT_CAPS + 255) / 256, 256, 0, stream>>>(s, v0, 0.1f);

    // iter 1: logits from v0
    zero_kernel<<<(SV_ELEMS + 255) / 256, 256, 0, stream>>>(s, SV_ELEMS);
    routing_iter<<<rout_grid, 128, 0, stream>>>(x, W, v0, 1, s);
    squash_kernel<<<(B_SZ * OUT_CAPS + 255) / 256, 256, 0, stream>>>(s, v1, 1.0f);

    // iter 2: logits from v0 + v1, final v -> d_out
    zero_kernel<<<(SV_ELEMS + 255) / 256, 256, 0, stream>>>(s, SV_ELEMS);
    routing_iter<<<rout_grid, 128, 0, stream>>>(x, W, v0, 2, s);
    squash_kernel<<<(B_SZ * OUT_CAPS + 255) / 256, 256, 0, stream>>>(s, out, 1.0f);
}